// MTPWorker_12214886990264
// MI455X (gfx1250) — compile-verified
//
#include <hip/hip_runtime.h>
#include <stdint.h>

#define BB 64
#define KK 3
#define HH 4096
#define H2 8192
#define VV 32000

// d_out flat layout (floats): accepted_tokens[256] | num_accepted[64] |
// next_draft_tokens[192] | next_new_tokens[256] | new_past_t[192]  => 960
#define OUT_ACC 0
#define OUT_NA  256
#define OUT_NDT 320
#define OUT_NNT 512
#define OUT_NPT 768

typedef float v2f __attribute__((ext_vector_type(2)));
typedef float v8f __attribute__((ext_vector_type(8)));

// Monotonic mapping of fp32 to orderable u32 (larger float -> larger key).
__device__ __forceinline__ unsigned fkey(float f) {
  unsigned u = __float_as_uint(f);
  return (u & 0x80000000u) ? ~u : (u | 0x80000000u);
}

// pack (value, column) so u64 max == argmax with lowest-index tie-break
__device__ __forceinline__ unsigned long long packmax(float v, int col) {
  return ((unsigned long long)fkey(v) << 32) |
         (unsigned long long)(0x7FFFFFFFu - (unsigned)col);
}

__device__ __forceinline__ unsigned long long shfl_xor_u64(unsigned long long v, int m) {
  int lo = __shfl_xor((int)(v & 0xFFFFFFFFull), m, 32);
  int hi = __shfl_xor((int)(v >> 32), m, 32);
  return ((unsigned long long)(unsigned)hi << 32) | (unsigned)(unsigned int)lo;
}

// ---------------------------------------------------------------------------
// 1) argmax over logits rows: 256 rows x 32000 cols, one block per row
// ---------------------------------------------------------------------------
__global__ void k_argmax_logits(const float* __restrict__ logits,
                                int* __restrict__ target) {
  const int row = blockIdx.x;
  const float* p = logits + (size_t)row * VV;
  unsigned long long best = 0ull;
  for (int v = threadIdx.x; v < VV; v += 256) {
    unsigned long long pk = packmax(p[v], v);
    if (pk > best) best = pk;
  }
  __shared__ unsigned long long s[256];
  s[threadIdx.x] = best;
  __syncthreads();
  for (int off = 128; off > 0; off >>= 1) {
    if (threadIdx.x < off) {
      if (s[threadIdx.x + off] > s[threadIdx.x]) s[threadIdx.x] = s[threadIdx.x + off];
    }
    __syncthreads();
  }
  if (threadIdx.x == 0)
    target[row] = (int)(0x7FFFFFFFu - (unsigned)(s[0] & 0xFFFFFFFFull));
}

// ---------------------------------------------------------------------------
// 2) acceptance bookkeeping + gather h = hidden[b, num_accepted-1]
//    also zeroes the per-iteration argmax accumulators (graph-replay safe)
// ---------------------------------------------------------------------------
__global__ void k_accept(const int* __restrict__ target,
                         const int* __restrict__ draft,
                         const int* __restrict__ past_t,
                         const int* __restrict__ slot,
                         const float* __restrict__ hidden,
                         int* __restrict__ tok_ws,
                         unsigned long long* __restrict__ pk,
                         float* __restrict__ hbuf,
                         float* __restrict__ out) {
  const int b = blockIdx.x;
  __shared__ int s_na;
  if (threadIdx.x == 0) {
    int t[KK + 1];
#pragma unroll
    for (int i = 0; i <= KK; ++i) t[i] = target[b * (KK + 1) + i];
    int na = 1, run = 1;
#pragma unroll
    for (int i = 0; i < KK; ++i) {
      run &= (draft[b * KK + i] == t[i]) ? 1 : 0;
      na += run;
    }
#pragma unroll
    for (int i = 0; i <= KK; ++i) out[OUT_ACC + b * (KK + 1) + i] = (float)t[i];
    out[OUT_NA + b] = (float)na;
    int seq[2 * KK + 1];
    const int sb = slot[b];
#pragma unroll
    for (int i = 0; i < KK; ++i) seq[i] = past_t[sb * KK + i];
#pragma unroll
    for (int i = 0; i <= KK; ++i) seq[KK + i] = t[i];
#pragma unroll
    for (int i = 0; i < KK; ++i) out[OUT_NPT + b * KK + i] = (float)seq[na + i];
    const int tok0 = t[na - 1];
    tok_ws[b] = tok0;
    out[OUT_NNT + b * (KK + 1) + 0] = (float)tok0;
    s_na = na;
  }
  if (b == 0 && threadIdx.x < KK * BB) pk[threadIdx.x] = 0ull;  // zero accumulators
  __syncthreads();
  const int na = s_na;
  const float* src = hidden + (size_t)(b * (KK + 1) + (na - 1)) * HH;
  for (int i = threadIdx.x; i < HH; i += blockDim.x) hbuf[(size_t)b * HH + i] = src[i];
}

// ---------------------------------------------------------------------------
// 3) build xin[b] = [ rmsnorm(embed[tok[b]]) , rmsnorm(h[b]) ]   (B x 2H)
// ---------------------------------------------------------------------------
__global__ void k_rms_xin(const float* __restrict__ embed,
                          const int* __restrict__ tok_ws,
                          const float* __restrict__ h,
                          float* __restrict__ xin) {
  const int b = blockIdx.x;
  const int half = blockIdx.y;  // 0 = embed row, 1 = hidden row
  const float* src = (half == 0) ? (embed + (size_t)tok_ws[b] * HH)
                                 : (h + (size_t)b * HH);
  float ss = 0.f;
  for (int i = threadIdx.x; i < HH; i += 256) {
    float v = src[i];
    ss += v * v;
  }
  __shared__ float s[256];
  s[threadIdx.x] = ss;
  __syncthreads();
  for (int off = 128; off > 0; off >>= 1) {
    if (threadIdx.x < off) s[threadIdx.x] += s[threadIdx.x + off];
    __syncthreads();
  }
  const float scale = rsqrtf(s[0] / (float)HH + 1e-6f);
  float* dst = xin + (size_t)b * H2 + (size_t)half * HH;
  for (int i = threadIdx.x; i < HH; i += 256) dst[i] = src[i] * scale;
}

// ---------------------------------------------------------------------------
// 4) FC GEMM: xout[64 x 4096] = xin[64 x 8192] * W[8192 x 4096]
//    One 128-thread block (4 waves) per 16-column N tile: wave w = M-tile w.
//    All 4 waves stream the SAME weight tile -> WGP$ reuse, ~1x HBM traffic.
// ---------------------------------------------------------------------------
__global__ void __launch_bounds__(128)
k_gemm_fc(const float* __restrict__ xin,
          const float* __restrict__ w,
          float* __restrict__ xout) {
  const int lane = threadIdx.x & 31;
  const int mbase = (threadIdx.x >> 5) * 16;  // 4 waves cover M = 64
  const int nbase = blockIdx.x * 16;          // N tile (H/16)
  const int nl = lane & 15;
  const int k0 = (lane >> 4) * 2;             // A: lanes 0-15 K{0,1}, 16-31 K{2,3}
  const float* arow = xin + (size_t)(mbase + nl) * H2 + k0;
  const float* bcol = w + (size_t)k0 * HH + (nbase + nl);
  v8f c = {};
  for (int kb = 0; kb < H2; kb += 32) {
    if (kb + 128 < H2)  // prefetch weight rows ~128 K-steps ahead
      __builtin_prefetch(bcol + (size_t)(kb + 128) * HH, 0, 0);
#pragma unroll
    for (int kk = 0; kk < 32; kk += 4) {
      const int k = kb + kk;
      v2f a = *reinterpret_cast<const v2f*>(arow + k);
      v2f bb;
      bb.x = bcol[(size_t)k * HH];
      bb.y = bcol[(size_t)(k + 1) * HH];
      c = __builtin_amdgcn_wmma_f32_16x16x4_f32(false, a, false, bb,
                                                (short)0, c, false, false);
    }
  }
  const int col = nbase + nl;
  const int rbase = mbase + ((lane < 16) ? 0 : 8);
#pragma unroll
  for (int r = 0; r < 8; ++r) xout[(size_t)(rbase + r) * HH + col] = c[r];
}

// ---------------------------------------------------------------------------
// 5) lm_head GEMM + fused argmax: dlog[64 x 32000] = x[64x4096] * lm_head^T
//    One 128-thread block (4 waves) per 16-row V tile; the 524 MB lm_head
//    stream is read once (4 waves share it through WGP$). Each lane streams
//    a contiguous row of lm_head; prefetch 4 KB ahead.
// ---------------------------------------------------------------------------
__global__ void __launch_bounds__(128)
k_gemm_head_argmax(const float* __restrict__ x,
                   const float* __restrict__ lm,
                   unsigned long long* __restrict__ pk) {
  const int lane = threadIdx.x & 31;
  const int mbase = (threadIdx.x >> 5) * 16;  // 4 waves cover M = 64
  const int vbase = blockIdx.x * 16;          // V tile (V/16)
  const int nl = lane & 15;
  const int k0 = (lane >> 4) * 2;
  const float* arow = x + (size_t)(mbase + nl) * HH + k0;
  const float* brow = lm + (size_t)(vbase + nl) * HH + k0;
  v8f c = {};
  for (int kb = 0; kb < HH; kb += 32) {
    if (kb + 1024 < HH)  // prefetch this lane's weight row 4 KB ahead
      __builtin_prefetch(brow + kb + 1024, 0, 0);
#pragma unroll
    for (int kk = 0; kk < 32; kk += 4) {
      const int k = kb + kk;
      v2f a = *reinterpret_cast<const v2f*>(arow + k);
      v2f b = *reinterpret_cast<const v2f*>(brow + k);
      c = __builtin_amdgcn_wmma_f32_16x16x4_f32(false, a, false, b,
                                                (short)0, c, false, false);
    }
  }
  const int col = vbase + nl;
  const int rbase = mbase + ((lane < 16) ? 0 : 8);
#pragma unroll
  for (int r = 0; r < 8; ++r) {
    unsigned long long p = packmax(c[r], col);
    // reduce over the 16 lanes (one column each) that share this row
#pragma unroll
    for (int m = 1; m < 16; m <<= 1) {
      unsigned long long o = shfl_xor_u64(p, m);
      if (o > p) p = o;
    }
    if (nl == 0) atomicMax(&pk[rbase + r], p);
  }
}

// ---------------------------------------------------------------------------
// 6) decode per-row argmax -> next token, write outputs
// ---------------------------------------------------------------------------
__global__ void k_decode(const unsigned long long* __restrict__ pk,
                         int* __restrict__ tok_ws,
                         float* __restrict__ out, int j) {
  const int b = threadIdx.x;
  if (b < BB) {
    const int v = (int)(0x7FFFFFFFu - (unsigned)(pk[b] & 0xFFFFFFFFull));
    tok_ws[b] = v;
    out[OUT_NDT + b * KK + j] = (float)v;
    out[OUT_NNT + b * (KK + 1) + 1 + j] = (float)v;
  }
}

// ---------------------------------------------------------------------------
extern "C" void kernel_launch(void* const* d_in, const int* in_sizes, int n_in,
                              void* d_out, int out_size, void* d_ws, size_t ws_size,
                              hipStream_t stream) {
  (void)in_sizes; (void)n_in; (void)out_size; (void)ws_size;
  const float* logits  = (const float*)d_in[0];
  const float* hidden  = (const float*)d_in[1];
  const float* past_h  = (const float*)d_in[2]; (void)past_h; // never observable
  const float* embed   = (const float*)d_in[3];
  const float* lm_head = (const float*)d_in[4];
  const float* mtp_fc  = (const float*)d_in[5];
  const int*   draft   = (const int*)d_in[6];
  const int*   past_t  = (const int*)d_in[7];
  const int*   slot    = (const int*)d_in[8];
  float* out = (float*)d_out;

  char* ws = (char*)d_ws;
  int* target  = (int*)(ws + 0);                       // 256 ints
  int* tok_ws  = (int*)(ws + 1024);                    // 64 ints
  unsigned long long* pk = (unsigned long long*)(ws + 1536); // 3*64 u64
  float* hbuf  = (float*)(ws + 4096);                  // 64*4096
  float* xout  = (float*)(ws + 4096 + (size_t)BB * HH * 4);
  float* xin   = (float*)(ws + 4096 + (size_t)2 * BB * HH * 4); // 64*8192

  k_argmax_logits<<<BB * (KK + 1), 256, 0, stream>>>(logits, target);
  k_accept<<<BB, 256, 0, stream>>>(target, draft, past_t, slot, hidden,
                                   tok_ws, pk, hbuf, out);

  float* hcur = hbuf;
  float* xcur = xout;
  for (int j = 0; j < KK; ++j) {
    k_rms_xin<<<dim3(BB, 2), 256, 0, stream>>>(embed, tok_ws, hcur, xin);
    k_gemm_fc<<<HH / 16, 128, 0, stream>>>(
        xin, mtp_fc + (size_t)j * H2 * HH, xcur);
    k_gemm_head_argmax<<<VV / 16, 128, 0, stream>>>(
        xcur, lm_head, pk + (size_t)j * BB);
    k_decode<<<1, 64, 0, stream>>>(pk + (size_t)j * BB, tok_ws, out, j);
    float* t = hcur; hcur = xcur; xcur = t;  // x becomes next h
  }
}